// KmeansModel_70368744178459
// MI455X (gfx1250) — compile-verified
//
#include <hip/hip_runtime.h>

typedef __attribute__((ext_vector_type(2))) float v2f;
typedef __attribute__((ext_vector_type(4))) float v4f;
typedef __attribute__((ext_vector_type(8))) float v8f;
typedef __attribute__((ext_vector_type(4))) int   v4i;

#if __has_builtin(__builtin_amdgcn_global_load_async_to_lds_b128)
#define KM_ASYNC 1
#else
#define KM_ASYNC 0
#endif

// Problem constants (fixed by the reference setup)
constexpr int N = 32768;
constexpr int D = 1024;
constexpr int K = 8192;

// Tiling: 256 threads = 8 waves; block tile 128 rows x 128 centroids,
// full-D register accumulation; d staged in double-buffered 16-wide chunks.
constexpr int BM  = 128;       // rows per block
constexpr int BK  = 128;       // centroid columns per k-chunk
constexpr int DC  = 16;        // D-chunk staged in LDS per pipeline stage
constexpr int NCH = D / DC;    // 64 chunks
constexpr int LXS = 20;        // X LDS row stride: 80B rows, 16B aligned, conflict-free
constexpr int LCS = 132;       // C LDS row stride: 528B rows, 16B aligned

// ---------------------------------------------------------------------------
// Kernel 1: half centroid norms  hcn[k] = 0.5 * sum_d C[d,k]^2   (C is [D,K])
// ---------------------------------------------------------------------------
__global__ __launch_bounds__(256)
void kmeans_cnorm_kernel(const float* __restrict__ Cm, float* __restrict__ hcn) {
    const int k = blockIdx.x * 256 + threadIdx.x;   // coalesced across k
    float s = 0.0f;
    #pragma unroll 8
    for (int d = 0; d < D; ++d) {
        const float c = Cm[(size_t)d * K + k];
        s = fmaf(c, c, s);
    }
    hcn[k] = 0.5f * s;
}

// ---------------------------------------------------------------------------
// Stage one (X 128x16, C 16x128) chunk into an LDS buffer.
// Async path: GLOBAL_LOAD_ASYNC_TO_LDS_B128 (ASYNCcnt), 4 issues per wave.
// Fallback: synchronous b128 load + ds_store (DScnt handled by barrier).
// ---------------------------------------------------------------------------
__device__ __forceinline__
void km_stage(const float* __restrict__ X, const float* __restrict__ Cm,
              float* __restrict__ lxb, float* __restrict__ lcb,
              int t, int row0, int kb, int db) {
#if KM_ASYNC
    #pragma unroll
    for (int i = 0; i < 2; ++i) {               // X: 512 float4 slots
        const int s = t + i * 256;
        const int r = s >> 2, c4 = (s & 3) << 2;
        __builtin_amdgcn_global_load_async_to_lds_b128(
            (v4i*)&X[(size_t)(row0 + r) * D + db + c4],
            (v4i*)&lxb[r * LXS + c4], 0, 0);
    }
    #pragma unroll
    for (int i = 0; i < 2; ++i) {               // C: 512 float4 slots
        const int s = t + i * 256;
        const int dr = s >> 5, k4 = (s & 31) << 2;
        __builtin_amdgcn_global_load_async_to_lds_b128(
            (v4i*)&Cm[(size_t)(db + dr) * K + kb + k4],
            (v4i*)&lcb[dr * LCS + k4], 0, 0);
    }
#else
    #pragma unroll
    for (int i = 0; i < 2; ++i) {
        const int s = t + i * 256;
        const int r = s >> 2, c4 = (s & 3) << 2;
        *(v4f*)&lxb[r * LXS + c4] = *(const v4f*)&X[(size_t)(row0 + r) * D + db + c4];
    }
    #pragma unroll
    for (int i = 0; i < 2; ++i) {
        const int s = t + i * 256;
        const int dr = s >> 5, k4 = (s & 31) << 2;
        *(v4f*)&lcb[dr * LCS + k4] = *(const v4f*)&Cm[(size_t)(db + dr) * K + kb + k4];
    }
#endif
}

// ---------------------------------------------------------------------------
// Kernel 2: fused GEMM (x @ C) + argmax_k (dot - 0.5*||c_k||^2) epilogue
// using V_WMMA_F32_16X16X4_F32 with async double-buffered LDS staging.
// ---------------------------------------------------------------------------
__global__ __launch_bounds__(256)
void kmeans_assign_wmma_kernel(const float* __restrict__ X,
                               const float* __restrict__ Cm,
                               const float* __restrict__ hcn,
                               float* __restrict__ out) {
    __shared__ float lx[2][BM * LXS];   // 2 x 128x16 x-tiles (padded)
    __shared__ float lc[2][DC * LCS];   // 2 x 16x128 centroid tiles (padded)

    const int t    = threadIdx.x;
    const int lane = t & 31;
    const int w    = t >> 5;        // wave 0..7
    const int wm   = w >> 1;        // wave row group 0..3  (32 rows each)
    const int wn   = w & 1;         // wave col group 0..1  (64 cols each)
    const int lh   = lane >> 4;     // lane half 0/1
    const int l15  = lane & 15;
    const int row0 = blockIdx.x * BM;

    // Per-lane running best over k for the 16 matrix rows this lane owns:
    // j = im*8 + v  ->  row (wm*32 + im*16 + 8*lh + v)
    float bv[16];
    int   bi[16];
    #pragma unroll
    for (int j = 0; j < 16; ++j) { bv[j] = -3.402823466e38f; bi[j] = 0; }

    for (int kb = 0; kb < K; kb += BK) {
        v8f acc[2][4];
        #pragma unroll
        for (int im = 0; im < 2; ++im)
            #pragma unroll
            for (int in = 0; in < 4; ++in)
                acc[im][in] = 0.0f;

        // Prologue: stage chunk 0 (previous k-chunk's trailing barrier
        // guarantees both buffers are free here).
        km_stage(X, Cm, lx[0], lc[0], t, row0, kb, 0);

        for (int ci = 0; ci < NCH; ++ci) {
            const int cur = ci & 1;

            if (ci + 1 < NCH) {
                // Issue next chunk into the spare buffer, then wait for the
                // 4 older (current-buffer) async loads of this wave.
                km_stage(X, Cm, lx[cur ^ 1], lc[cur ^ 1], t, row0, kb, (ci + 1) * DC);
#if KM_ASYNC
                __asm__ volatile("s_wait_asynccnt 0x4" ::: "memory");
#endif
            }
#if KM_ASYNC
            else {
                __asm__ volatile("s_wait_asynccnt 0x0" ::: "memory");
            }
#endif
            __syncthreads();   // all waves' loads for `cur` have landed

            const float* __restrict__ lxb = lx[cur];
            const float* __restrict__ lcb = lc[cur];

            // 4 k-steps of 4; per step: 2 A-frags, 4 B-frags, 8 WMMAs
            #pragma unroll
            for (int kk = 0; kk < DC; kk += 4) {
                const int ca = kk + 2 * lh;   // A/B K-offset per lane half
                // A 16x4 fp32 frag: lane m=l15; VGPR0=K(ca), VGPR1=K(ca+1)
                const v2f a0 = *(const v2f*)&lxb[(wm * 32 +      l15) * LXS + ca];
                const v2f a1 = *(const v2f*)&lxb[(wm * 32 + 16 + l15) * LXS + ca];
                #pragma unroll
                for (int in = 0; in < 4; ++in) {
                    const int n = wn * 64 + in * 16 + l15;
                    v2f b;
                    b.x = lcb[ ca      * LCS + n];
                    b.y = lcb[(ca + 1) * LCS + n];
                    acc[0][in] = __builtin_amdgcn_wmma_f32_16x16x4_f32(
                        false, a0, false, b, (short)0, acc[0][in], false, false);
                    acc[1][in] = __builtin_amdgcn_wmma_f32_16x16x4_f32(
                        false, a1, false, b, (short)0, acc[1][in], false, false);
                }
            }
            __syncthreads();   // release `cur` before it is re-staged
        }

        // Epilogue: score = dot - 0.5||c||^2 ; running argmax (ties -> lower k)
        #pragma unroll
        for (int in = 0; in < 4; ++in) {
            const int   kidx = kb + wn * 64 + in * 16 + l15;
            const float hc   = hcn[kidx];
            #pragma unroll
            for (int im = 0; im < 2; ++im) {
                #pragma unroll
                for (int v = 0; v < 8; ++v) {
                    const float s = acc[im][in][v] - hc;
                    const int   j = im * 8 + v;
                    if (s > bv[j] || (s == bv[j] && kidx < bi[j])) {
                        bv[j] = s;
                        bi[j] = kidx;
                    }
                }
            }
        }
    }

    // Final reduction: each matrix row's candidates live in one 16-lane half.
    #pragma unroll
    for (int j = 0; j < 16; ++j) {
        float val = bv[j];
        int   idx = bi[j];
        #pragma unroll
        for (int off = 8; off; off >>= 1) {
            const float ov = __shfl_xor(val, off, 16);
            const int   oi = __shfl_xor(idx, off, 16);
            if (ov > val || (ov == val && oi < idx)) { val = ov; idx = oi; }
        }
        if (l15 == 0) {
            const int row = row0 + wm * 32 + (j >> 3) * 16 + 8 * lh + (j & 7);
            out[row] = (float)idx;   // indices <= 8191: exact in fp32
        }
    }
}

// ---------------------------------------------------------------------------
extern "C" void kernel_launch(void* const* d_in, const int* in_sizes, int n_in,
                              void* d_out, int out_size, void* d_ws, size_t ws_size,
                              hipStream_t stream) {
    const float* x   = (const float*)d_in[0];   // [N, D] fp32
    const float* cm  = (const float*)d_in[1];   // [D, K] fp32
    float*       out = (float*)d_out;           // [N] argmin indices
    float*       hcn = (float*)d_ws;            // [K] 0.5*||c_k||^2 scratch

    kmeans_cnorm_kernel<<<K / 256, 256, 0, stream>>>(cm, hcn);
    kmeans_assign_wmma_kernel<<<N / BM, 256, 0, stream>>>(x, cm, hcn, out);
}